// SpKBGATModified_38955353375000
// MI455X (gfx1250) — compile-verified
//
#include <hip/hip_runtime.h>
#include <math.h>

#define DD 128
#define ALPHA_LK 0.2f
#define EPS_SUM 1e-19f
#define NORM_EPS 1e-12f

typedef __attribute__((ext_vector_type(2))) float v2f;
typedef __attribute__((ext_vector_type(8))) float v8f;

// ---------------------------------------------------------------------------
// WMMA f32 GEMM: C[M x 128] = A[M x 128] @ op(B)
//   TRANSB=1: C[m,n] = sum_k A[m,k] * B[n*ldb + k]
//   TRANSB=0: C[m,n] = sum_k A[m,k] * B[k*ldb + n]
// One wave32 per 16x16 tile; 32 x v_wmma_f32_16x16x4_f32 over K=128.
// A frag (16x4 f32): lane<16 holds row=lane, k={0,1}; lane>=16 row=lane-16, k={2,3}
// B frag (4x16 f32): lane<16 holds col=lane, k={0,1}; lane>=16 col=lane-16, k={2,3}
// C frag (16x16 f32, 8 VGPRs): vgpr r, lane<16 -> (row r,     col lane)
//                                      lane>=16 -> (row r + 8, col lane-16)
// ---------------------------------------------------------------------------
template <int TRANSB>
__global__ __launch_bounds__(32)
void wmma_gemm_f32_kernel(const float* __restrict__ A, const float* __restrict__ B,
                          float* __restrict__ C, int M, int ldb)
{
    const int lane = threadIdx.x & 31;
    const int half = lane >> 4;
    const int l16  = lane & 15;
    const int rowTile = blockIdx.x * 16;
    const int colTile = blockIdx.y * 16;

    int arow = rowTile + l16;
    if (arow >= M) arow = M - 1;           // clamp loads (stores are guarded)
    const int bcol = colTile + l16;

    v8f acc = {};
    #pragma unroll
    for (int kb = 0; kb < DD; kb += 4) {
        const int k2 = kb + half * 2;
        v2f af = *(const v2f*)(A + (size_t)arow * DD + k2);
        v2f bf;
        if (TRANSB) {
            bf = *(const v2f*)(B + (size_t)bcol * ldb + k2);
        } else {
            bf[0] = B[(size_t)(k2 + 0) * ldb + bcol];
            bf[1] = B[(size_t)(k2 + 1) * ldb + bcol];
        }
        acc = __builtin_amdgcn_wmma_f32_16x16x4_f32(
            /*neg_a=*/false, af, /*neg_b=*/false, bf,
            /*c_mod=*/(short)0, acc, /*reuse_a=*/false, /*reuse_b=*/false);
    }

    #pragma unroll
    for (int r = 0; r < 8; ++r) {
        const int row = rowTile + r + half * 8;
        if (row < M) C[(size_t)row * DD + colTile + l16] = acc[r];
    }
}

// ---------------------------------------------------------------------------
// out[row] = dot(P[row, :], a2) ; one wave32 per row, 4 floats per lane
// ---------------------------------------------------------------------------
__device__ __forceinline__ float wave_allreduce_sum(float v) {
    #pragma unroll
    for (int off = 16; off > 0; off >>= 1) v += __shfl_xor(v, off, 32);
    return v;
}

__global__ __launch_bounds__(128)
void row_dot_kernel(const float* __restrict__ P, const float* __restrict__ a2,
                    float* __restrict__ out, int rows)
{
    const int lane = threadIdx.x & 31;
    const int wave = threadIdx.x >> 5;
    const int row  = blockIdx.x * 4 + wave;
    if (row >= rows) return;
    const float4 p = *(const float4*)(P + (size_t)row * DD + lane * 4);
    const float4 v = *(const float4*)(a2 + lane * 4);
    float s = p.x * v.x + p.y * v.y + p.z * v.z + p.w * v.w;
    s = wave_allreduce_sum(s);
    if (lane == 0) out[row] = s;
}

// ---------------------------------------------------------------------------
// zero accumulators (must run every call: harness does not re-poison)
// ---------------------------------------------------------------------------
__global__ void zero_kernel(float* __restrict__ p, long n)
{
    long i = (long)blockIdx.x * blockDim.x + threadIdx.x;
    const long stride = (long)gridDim.x * blockDim.x;
    for (; i < n; i += stride) p[i] = 0.0f;
}

// ---------------------------------------------------------------------------
// Edge phase: one wave32 per edge.
//   w = exp(-leaky_relu(pt_s[t] + ps_s[s] + pr_s[r], 0.2))
//   S[t]      += w                        (lane 0)
//   h[t, c]   += w * (Ps[s,c] + Pr[r,c])  (4 cols / lane, f32 atomics; L2-resident)
// ---------------------------------------------------------------------------
__global__ __launch_bounds__(256)
void edge_scatter_kernel(const int* __restrict__ tgt, const int* __restrict__ src,
                         const int* __restrict__ rl,
                         const float* __restrict__ pt_s, const float* __restrict__ ps_s,
                         const float* __restrict__ pr_s,
                         const float* __restrict__ Ps, const float* __restrict__ Pr,
                         float* __restrict__ S, float* __restrict__ h, int nE)
{
    const long gid = (long)blockIdx.x * blockDim.x + threadIdx.x;
    const int edge = (int)(gid >> 5);
    const int lane = (int)(gid & 31);
    if (edge >= nE) return;

    const int t = tgt[edge];
    const int s = src[edge];
    const int r = rl[edge];

    const float p  = pt_s[t] + ps_s[s] + pr_s[r];
    const float lr = p > 0.0f ? p : ALPHA_LK * p;
    const float w  = __expf(-lr);

    if (lane == 0) atomicAdd(S + t, w);

    const float4 a = *(const float4*)(Ps + (size_t)s * DD + lane * 4);
    const float4 b = *(const float4*)(Pr + (size_t)r * DD + lane * 4);
    float* hb = h + (size_t)t * DD + lane * 4;
    atomicAdd(hb + 0, w * (a.x + b.x));
    atomicAdd(hb + 1, w * (a.y + b.y));
    atomicAdd(hb + 2, w * (a.z + b.z));
    atomicAdd(hb + 3, w * (a.w + b.w));
}

// ---------------------------------------------------------------------------
// Entities: h_full = (h + Pt*S) / (S + EPS); gat = relu(elu(h_full)) = max(h_full,0)
//           o = Went + gat; out = o / max(||o||, 1e-12). One wave32 per row.
// ---------------------------------------------------------------------------
__global__ __launch_bounds__(256)
void finalize_ent_kernel(const float* __restrict__ h, const float* __restrict__ Pt,
                         const float* __restrict__ Went, const float* __restrict__ S,
                         float* __restrict__ out, int rows)
{
    const long gid = (long)blockIdx.x * blockDim.x + threadIdx.x;
    const int row  = (int)(gid >> 5);
    const int lane = (int)(gid & 31);
    if (row >= rows) return;

    const float Sv  = S[row];
    const float inv = 1.0f / (Sv + EPS_SUM);
    const size_t base = (size_t)row * DD + lane * 4;

    const float4 hp = *(const float4*)(h + base);
    const float4 pt = *(const float4*)(Pt + base);
    const float4 we = *(const float4*)(Went + base);

    float4 o;
    o.x = we.x + fmaxf((hp.x + pt.x * Sv) * inv, 0.0f);
    o.y = we.y + fmaxf((hp.y + pt.y * Sv) * inv, 0.0f);
    o.z = we.z + fmaxf((hp.z + pt.z * Sv) * inv, 0.0f);
    o.w = we.w + fmaxf((hp.w + pt.w * Sv) * inv, 0.0f);

    float sq = o.x * o.x + o.y * o.y + o.z * o.z + o.w * o.w;
    sq = wave_allreduce_sum(sq);
    const float scale = 1.0f / fmaxf(sqrtf(sq), NORM_EPS);

    float4 res = { o.x * scale, o.y * scale, o.z * scale, o.w * scale };
    *(float4*)(out + base) = res;
}

// ---------------------------------------------------------------------------
// Relations: o = relu(rel @ W_r) + rel; out = o / max(||o||, 1e-12)
// ---------------------------------------------------------------------------
__global__ __launch_bounds__(256)
void finalize_rel_kernel(const float* __restrict__ RelW, const float* __restrict__ rel,
                         float* __restrict__ out, int rows)
{
    const long gid = (long)blockIdx.x * blockDim.x + threadIdx.x;
    const int row  = (int)(gid >> 5);
    const int lane = (int)(gid & 31);
    if (row >= rows) return;

    const size_t base = (size_t)row * DD + lane * 4;
    const float4 rw = *(const float4*)(RelW + base);
    const float4 rr = *(const float4*)(rel + base);

    float4 o;
    o.x = fmaxf(rw.x, 0.0f) + rr.x;
    o.y = fmaxf(rw.y, 0.0f) + rr.y;
    o.z = fmaxf(rw.z, 0.0f) + rr.z;
    o.w = fmaxf(rw.w, 0.0f) + rr.w;

    float sq = o.x * o.x + o.y * o.y + o.z * o.z + o.w * o.w;
    sq = wave_allreduce_sum(sq);
    const float scale = 1.0f / fmaxf(sqrtf(sq), NORM_EPS);

    float4 res = { o.x * scale, o.y * scale, o.z * scale, o.w * scale };
    *(float4*)(out + base) = res;
}

// ---------------------------------------------------------------------------
extern "C" void kernel_launch(void* const* d_in, const int* in_sizes, int n_in,
                              void* d_out, int out_size, void* d_ws, size_t ws_size,
                              hipStream_t stream)
{
    const float* ent      = (const float*)d_in[0];   // (N, 128)
    const float* rel      = (const float*)d_in[1];   // (R, 128)
    const int*   edges    = (const int*)d_in[2];     // (2, E)
    const int*   rel_list = (const int*)d_in[3];     // (E,)
    const float* a        = (const float*)d_in[4];   // (128, 384)
    const float* a2       = (const float*)d_in[5];   // (128,)
    const float* W_r      = (const float*)d_in[6];   // (128, 128)
    const float* W_ent    = (const float*)d_in[7];   // (128, 128)

    const int N = in_sizes[0] / DD;
    const int R = in_sizes[1] / DD;
    const int E = in_sizes[3];
    const int* tgt = edges;
    const int* src = edges + E;

    // workspace carve-out (floats)
    float* ws = (float*)d_ws;
    size_t off = 0;
    float* Pt   = ws + off; off += (size_t)N * DD;   // ent @ A_t^T
    float* Ps   = ws + off; off += (size_t)N * DD;   // ent @ A_s^T
    float* Went = ws + off; off += (size_t)N * DD;   // ent @ W_entities
    float* Pr   = ws + off; off += (size_t)R * DD;   // rel @ A_r^T
    float* RelW = ws + off; off += (size_t)R * DD;   // rel @ W_r
    float* pt_s = ws + off; off += (size_t)N;
    float* ps_s = ws + off; off += (size_t)N;
    float* pr_s = ws + off; off += (size_t)R;
    float* S    = ws + off; off += (size_t)N;        // S and h contiguous -> one zero pass
    float* h    = ws + off; off += (size_t)N * DD;

    // 1) zero accumulators (S followed by h: N*(D+1) floats)
    zero_kernel<<<2048, 256, 0, stream>>>(S, (long)N * (DD + 1));

    // 2) WMMA f32 GEMMs
    dim3 blk(32);
    dim3 gN((unsigned)((N + 15) / 16), DD / 16);
    dim3 gR((unsigned)((R + 15) / 16), DD / 16);
    wmma_gemm_f32_kernel<1><<<gN, blk, 0, stream>>>(ent, a,          Pt,   N, 3 * DD);
    wmma_gemm_f32_kernel<1><<<gN, blk, 0, stream>>>(ent, a + DD,     Ps,   N, 3 * DD);
    wmma_gemm_f32_kernel<1><<<gR, blk, 0, stream>>>(rel, a + 2 * DD, Pr,   R, 3 * DD);
    wmma_gemm_f32_kernel<0><<<gN, blk, 0, stream>>>(ent, W_ent,      Went, N, DD);
    wmma_gemm_f32_kernel<0><<<gR, blk, 0, stream>>>(rel, W_r,        RelW, R, DD);

    // 3) attention scalars: *_s = P @ a2
    row_dot_kernel<<<(N + 3) / 4, 128, 0, stream>>>(Pt, a2, pt_s, N);
    row_dot_kernel<<<(N + 3) / 4, 128, 0, stream>>>(Ps, a2, ps_s, N);
    row_dot_kernel<<<(R + 3) / 4, 128, 0, stream>>>(Pr, a2, pr_s, R);

    // 4) edge phase: one wave32 per edge
    {
        long threads = (long)E * 32;
        long blocks  = (threads + 255) / 256;
        edge_scatter_kernel<<<(unsigned)blocks, 256, 0, stream>>>(
            tgt, src, rel_list, pt_s, ps_s, pr_s, Ps, Pr, S, h, E);
    }

    // 5) finalize + l2norm
    {
        long threads = (long)N * 32;
        finalize_ent_kernel<<<(unsigned)((threads + 255) / 256), 256, 0, stream>>>(
            h, Pt, Went, S, (float*)d_out, N);
    }
    {
        long threads = (long)R * 32;
        finalize_rel_kernel<<<(unsigned)((threads + 255) / 256), 256, 0, stream>>>(
            RelW, rel, (float*)d_out + (size_t)N * DD, R);
    }
}